// PhosphenePlacementAlgorithm_21474836480686
// MI455X (gfx1250) — compile-verified
//
#include <hip/hip_runtime.h>
#include <stdint.h>

// ---------------------------------------------------------------------------
// PhosphenePlacement for MI455X (gfx1250, wave32).
//   P1 rowstats : TDM (tensor_load_to_lds) streams 16x256 f32 tiles of
//                 logits/uniform into LDS; row max (VALU) then row sum of
//                 exp via V_WMMA_F32_16X16X4_F32 (exact f32 accumulate).
//   P2 hist1    : 16384-bin histogram of float bits>>16 of softmax values
//                 (values in (0,1] -> bits <= 0x3F800000, bin <= 16256).
//   P3 scan1    : find threshold bin (single thread, 16K iters) + patch sum.
//   P4 hist2    : 65536-bin histogram of low 16 bits within threshold bin.
//   P5 scan2    : exact 32-bit threshold + #ties needed.
//   P6 select   : bits>thr -> set mask bit + append index; bits==thr -> tie list.
//   P7 ties     : pick lowest-index ties (rank select, E is tiny).
//   P8 render   : one block per dot, 25x25 Gaussian splat, weight from the
//                 transposed mask bit, atomic f32 adds into the canvas.
// Inputs (128 MiB) fit in the 192 MB L2, so re-sweeps in P2/P4/P6 are cheap.
// ---------------------------------------------------------------------------

#define G        4096
#define GSHIFT   12
#define NTOT     (G * G)
#define KDOTS    16384
#define PATCH    25
#define HALF     12
#define INV2R2   (1.0f / 32.0f)      /* 1/(2*radius^2), radius=4 */
#define EQCAP    8192
#define TILE_R   16
#define TILE_C   256
#define SIGMOID1 0.7310585786300049f
#define SIGMOID0 0.5f

// workspace layout (bytes)
#define OFF_ROWMAX (0u)
#define OFF_ROWSUM (16u * 1024u)
#define OFF_HIST1  (32u * 1024u)     /* 16384 u32 = 64KB  */
#define OFF_HIST2  (128u * 1024u)    /* 65536 u32 = 256KB */
#define OFF_SCAL   (384u * 1024u)    /* scalars           */
#define OFF_SEL    (388u * 1024u)    /* KDOTS u32 = 64KB  */
#define OFF_EQ     (452u * 1024u)    /* EQCAP u32 = 32KB  */
#define OFF_BITMAP (512u * 1024u)    /* NTOT/8 = 2MB      */
#define WS_BYTES   (2560u * 1024u)
// scal indices: 0=b1 1=cnt_gt 2=thr_bits 3=need_eq 4=sel_count 5=eq_count
//               6=psum_bits 7=R

typedef uint32_t u32;
typedef u32   u32x4 __attribute__((ext_vector_type(4)));
typedef u32   u32x8 __attribute__((ext_vector_type(8)));
typedef float v2f   __attribute__((ext_vector_type(2)));
typedef float v8f   __attribute__((ext_vector_type(8)));

#if defined(__has_builtin)
#if __has_builtin(__builtin_amdgcn_wmma_f32_16x16x4_f32)
#define PPA_HAVE_WMMA_F32 1
#endif
#endif

// --- TDM: load a TILE_R x TILE_C f32 tile (row-major, tensor stride G) to LDS
__device__ __forceinline__ void ppa_tdm_load_tile(u32 lds_off, const void* gsrc) {
  u32x4 g0;
  u32x8 g1;
  unsigned long long ga = (unsigned long long)gsrc;
  g0[0] = 1u;                                              // count=1 (valid D#)
  g0[1] = lds_off;                                         // lds_addr (bytes)
  g0[2] = (u32)ga;                                         // global_addr[31:0]
  g0[3] = (u32)((ga >> 32) & 0x01FFFFFFull) | (2u << 30);  // addr[56:32], type=2
  g1[0] = 0x00020000u;                    // data_size=2 (4B), no multicast
  g1[1] = ((u32)G & 0xFFFFu) << 16;       // tensor_dim0[15:0]
  g1[2] = ((u32)G & 0xFFFFu) << 16;       // dim0[31:16]=0 | tensor_dim1[15:0]
  g1[3] = ((u32)TILE_C) << 16;            // dim1[31:16]=0 | tile_dim0
  g1[4] = (u32)TILE_R;                    // tile_dim1 | tile_dim2=0
  g1[5] = (u32)G;                         // tensor_dim0_stride[31:0]
  g1[6] = 0u;
  g1[7] = 0u;
  asm volatile("tensor_load_to_lds %0, %1" ::"s"(g0), "s"(g1) : "memory");
}

__device__ __forceinline__ float ppa_softval(int i, const float* __restrict__ lg,
                                             const float* __restrict__ un,
                                             const float* __restrict__ rmax,
                                             const float* __restrict__ rsum) {
  int r = i >> GSHIFT;
  float y = lg[i] - logf(-logf(un[i]));
  return expf(y - rmax[r]) / rsum[r];
}

// --- zero helper ------------------------------------------------------------
__global__ void ppa_zero_kernel(u32* __restrict__ p, int n) {
  int st = gridDim.x * blockDim.x;
  for (int i = blockIdx.x * blockDim.x + threadIdx.x; i < n; i += st) p[i] = 0u;
}

// --- P1: row max + row sum(exp) --------------------------------------------
__global__ __launch_bounds__(32) void ppa_rowstats_kernel(
    const float* __restrict__ lg, const float* __restrict__ un,
    float* __restrict__ rmax, float* __restrict__ rsum) {
  __shared__ float tl[TILE_R * TILE_C];
  __shared__ float tu[TILE_R * TILE_C];
  const int lane = threadIdx.x;          // wave32: one wave per block
  const int r = lane & 15;
  const int kq = lane >> 4;              // 0 or 1 -> K column pair per WMMA layout
  const int rowbase = blockIdx.x * TILE_R;
  const u32 ltl = (u32)(uintptr_t)&tl[0];
  const u32 ltu = (u32)(uintptr_t)&tu[0];

  // sweep 1: row maxima of y = logit - log(-log(u))
  float mx = -3.4e38f;
  for (int ch = 0; ch < G / TILE_C; ++ch) {
    asm volatile("s_wait_dscnt 0x0" ::: "memory");  // LDS reads done before reuse
    ppa_tdm_load_tile(ltl, lg + (size_t)rowbase * G + ch * TILE_C);
    ppa_tdm_load_tile(ltu, un + (size_t)rowbase * G + ch * TILE_C);
    __builtin_amdgcn_s_wait_tensorcnt(0);
    for (int cc = 0; cc < TILE_C; cc += 4) {
      int c0 = cc + 2 * kq;
      float y0 = tl[r * TILE_C + c0] - logf(-logf(tu[r * TILE_C + c0]));
      float y1 = tl[r * TILE_C + c0 + 1] - logf(-logf(tu[r * TILE_C + c0 + 1]));
      mx = fmaxf(mx, fmaxf(y0, y1));
    }
  }
  mx = fmaxf(mx, __shfl_xor(mx, 16));  // lanes r and r+16 both hold row-r max

  // sweep 2: row sums of exp(y - max), accumulated on the matrix pipe
#ifdef PPA_HAVE_WMMA_F32
  v8f acc = {};
  const v2f bones = {1.0f, 1.0f};   // B = all-ones 4x16 -> D[m][n] = sum_k A[m][k]
#else
  float sacc = 0.0f;
#endif
  for (int ch = 0; ch < G / TILE_C; ++ch) {
    asm volatile("s_wait_dscnt 0x0" ::: "memory");
    ppa_tdm_load_tile(ltl, lg + (size_t)rowbase * G + ch * TILE_C);
    ppa_tdm_load_tile(ltu, un + (size_t)rowbase * G + ch * TILE_C);
    __builtin_amdgcn_s_wait_tensorcnt(0);
    for (int cc = 0; cc < TILE_C; cc += 4) {
      int c0 = cc + 2 * kq;
      float y0 = tl[r * TILE_C + c0] - logf(-logf(tu[r * TILE_C + c0]));
      float y1 = tl[r * TILE_C + c0 + 1] - logf(-logf(tu[r * TILE_C + c0 + 1]));
      float e0 = expf(y0 - mx);
      float e1 = expf(y1 - mx);
#ifdef PPA_HAVE_WMMA_F32
      v2f a;               // A 16x4 f32: lane m -> K0,K1 ; lane m+16 -> K2,K3
      a.x = e0;
      a.y = e1;
      acc = __builtin_amdgcn_wmma_f32_16x16x4_f32(false, a, false, bones,
                                                  (short)0, acc, false, false);
#else
      sacc += e0 + e1;
#endif
    }
  }
#ifdef PPA_HAVE_WMMA_F32
  // D layout: lane0 vgpr j = D[M=j][N=0] ; lane16 vgpr j = D[M=8+j][N=0]
  if (lane == 0) {
#pragma unroll
    for (int j = 0; j < 8; ++j) rsum[rowbase + j] = acc[j];
  }
  if (lane == 16) {
#pragma unroll
    for (int j = 0; j < 8; ++j) rsum[rowbase + 8 + j] = acc[j];
  }
#else
  sacc += __shfl_xor(sacc, 16);
  if (lane < 16) rsum[rowbase + lane] = sacc;
#endif
  if (lane < 16) rmax[rowbase + lane] = mx;
}

// --- P2: coarse histogram (bits >> 16), LDS-staged --------------------------
__global__ __launch_bounds__(256) void ppa_hist1_kernel(
    const float* __restrict__ lg, const float* __restrict__ un,
    const float* __restrict__ rmax, const float* __restrict__ rsum,
    u32* __restrict__ gh) {
  __shared__ u32 lh[16384];  // 64KB of the 320KB WGP LDS
  for (int j = threadIdx.x; j < 16384; j += 256) lh[j] = 0u;
  __syncthreads();
  int st = gridDim.x * blockDim.x;
  for (int i = blockIdx.x * blockDim.x + threadIdx.x; i < NTOT; i += st) {
    u32 b = __float_as_uint(ppa_softval(i, lg, un, rmax, rsum)) >> 16;
    if (b > 16383u) b = 16383u;
    atomicAdd(&lh[b], 1u);
  }
  __syncthreads();
  for (int j = threadIdx.x; j < 16384; j += 256) {
    u32 v = lh[j];
    if (v) atomicAdd(&gh[j], v);
  }
}

// --- P3: pick threshold bin, and compute the patch normalizer ---------------
__global__ void ppa_scan1_kernel(const u32* __restrict__ gh, u32* __restrict__ scal) {
  if (threadIdx.x != 0 || blockIdx.x != 0) return;
  u32 acc = 0, t = 0;
  for (int b = 16383; b >= 0; --b) {
    u32 h = gh[b];
    if (acc + h >= (u32)KDOTS) { t = (u32)b; break; }
    acc += h;
  }
  scal[0] = t;
  scal[1] = acc;
  scal[7] = (u32)KDOTS - acc;  // R: still needed from bin t
  float ps = 0.0f;
  for (int r = 0; r < PATCH; ++r)
    for (int c = 0; c < PATCH; ++c) {
      float dx = (float)(r - HALF), dy = (float)(c - HALF);
      ps += expf(-(dx * dx + dy * dy) * INV2R2);
    }
  scal[6] = __float_as_uint(ps);
}

// --- P4: fine histogram of low 16 bits inside the threshold bin -------------
__global__ __launch_bounds__(256) void ppa_hist2_kernel(
    const float* __restrict__ lg, const float* __restrict__ un,
    const float* __restrict__ rmax, const float* __restrict__ rsum,
    const u32* __restrict__ scal, u32* __restrict__ gh2) {
  u32 b1 = scal[0];
  int st = gridDim.x * blockDim.x;
  for (int i = blockIdx.x * blockDim.x + threadIdx.x; i < NTOT; i += st) {
    u32 bits = __float_as_uint(ppa_softval(i, lg, un, rmax, rsum));
    if ((bits >> 16) == b1) atomicAdd(&gh2[bits & 0xFFFFu], 1u);
  }
}

// --- P5: exact 32-bit threshold --------------------------------------------
__global__ void ppa_scan2_kernel(const u32* __restrict__ gh2, u32* __restrict__ scal) {
  if (threadIdx.x != 0 || blockIdx.x != 0) return;
  u32 R = scal[7], b1 = scal[0];
  u32 acc = 0, l = 0;
  for (int v = 65535; v >= 0; --v) {
    u32 h = gh2[v];
    if (acc + h >= R) { l = (u32)v; break; }
    acc += h;
  }
  scal[2] = (b1 << 16) | l;  // threshold bit pattern
  scal[3] = R - acc;         // ties needed at exactly the threshold value
}

// --- P6: build mask bitmap + selected-index list ----------------------------
__global__ __launch_bounds__(256) void ppa_select_kernel(
    const float* __restrict__ lg, const float* __restrict__ un,
    const float* __restrict__ rmax, const float* __restrict__ rsum,
    u32* __restrict__ scal, u32* __restrict__ bm, u32* __restrict__ sel,
    u32* __restrict__ eq) {
  u32 thr = scal[2];
  int st = gridDim.x * blockDim.x;
  for (int i = blockIdx.x * blockDim.x + threadIdx.x; i < NTOT; i += st) {
    u32 bits = __float_as_uint(ppa_softval(i, lg, un, rmax, rsum));
    if (bits > thr) {
      atomicOr(&bm[(u32)i >> 5], 1u << ((u32)i & 31u));
      u32 p = atomicAdd(&scal[4], 1u);
      if (p < (u32)KDOTS) sel[p] = (u32)i;
    } else if (bits == thr) {
      u32 e = atomicAdd(&scal[5], 1u);
      if (e < (u32)EQCAP) eq[e] = (u32)i;
    }
  }
}

// --- P7: lowest-index tie break ---------------------------------------------
__global__ __launch_bounds__(256) void ppa_ties_kernel(u32* __restrict__ scal,
                                                       u32* __restrict__ bm,
                                                       u32* __restrict__ sel,
                                                       const u32* __restrict__ eq) {
  u32 E = scal[5];
  if (E > (u32)EQCAP) E = (u32)EQCAP;
  u32 R2 = scal[3];
  for (u32 t = threadIdx.x; t < E; t += blockDim.x) {
    u32 v = eq[t];
    u32 rank = 0;
    for (u32 j = 0; j < E; ++j) rank += (eq[j] < v) ? 1u : 0u;
    if (rank < R2) {
      atomicOr(&bm[v >> 5], 1u << (v & 31u));
      u32 p = atomicAdd(&scal[4], 1u);
      if (p < (u32)KDOTS) sel[p] = v;
    }
  }
}

// --- P8: splat 25x25 Gaussians ----------------------------------------------
__global__ __launch_bounds__(128) void ppa_render_kernel(
    const u32* __restrict__ sel, const u32* __restrict__ scal,
    const u32* __restrict__ bm, float* __restrict__ out) {
  u32 nsel = scal[4];
  if (nsel > (u32)KDOTS) nsel = (u32)KDOTS;
  u32 d = blockIdx.x;
  if (d >= nsel) return;
  u32 idx = sel[d];
  int yi = (int)(idx >> GSHIFT);
  int xi = (int)(idx & (G - 1));
  u32 tf = ((u32)xi << GSHIFT) | (u32)yi;  // transposed flat index
  bool on = (bm[tf >> 5] >> (tf & 31u)) & 1u;
  float w = on ? SIGMOID1 : SIGMOID0;
  float scale = w / __uint_as_float(scal[6]);
  int pt = yi - HALF; if (pt < 0) pt = 0;
  int pl = xi - HALF; if (pl < 0) pl = 0;
  for (int t = threadIdx.x; t < PATCH * PATCH; t += blockDim.x) {
    int rr = t / PATCH, cc = t - rr * PATCH;
    int row = pt + rr, col = pl + cc;
    if (row < G && col < G) {
      float dx = (float)(rr - HALF), dy = (float)(cc - HALF);
      atomicAdd(&out[row * G + col], scale * expf(-(dx * dx + dy * dy) * INV2R2));
    }
  }
}

extern "C" void kernel_launch(void* const* d_in, const int* in_sizes, int n_in,
                              void* d_out, int out_size, void* d_ws, size_t ws_size,
                              hipStream_t stream) {
  (void)in_sizes; (void)n_in; (void)out_size; (void)ws_size;
  const float* lg = (const float*)d_in[0];
  const float* un = (const float*)d_in[1];
  float* out = (float*)d_out;
  char* ws = (char*)d_ws;
  float* rmax = (float*)(ws + OFF_ROWMAX);
  float* rsum = (float*)(ws + OFF_ROWSUM);
  u32* h1 = (u32*)(ws + OFF_HIST1);
  u32* h2 = (u32*)(ws + OFF_HIST2);
  u32* scal = (u32*)(ws + OFF_SCAL);
  u32* sel = (u32*)(ws + OFF_SEL);
  u32* eq = (u32*)(ws + OFF_EQ);
  u32* bm = (u32*)(ws + OFF_BITMAP);

  ppa_zero_kernel<<<1024, 256, 0, stream>>>((u32*)ws, (int)(WS_BYTES / 4u));
  ppa_zero_kernel<<<4096, 256, 0, stream>>>((u32*)out, NTOT);
  ppa_rowstats_kernel<<<G / TILE_R, 32, 0, stream>>>(lg, un, rmax, rsum);
  ppa_hist1_kernel<<<2048, 256, 0, stream>>>(lg, un, rmax, rsum, h1);
  ppa_scan1_kernel<<<1, 1, 0, stream>>>(h1, scal);
  ppa_hist2_kernel<<<2048, 256, 0, stream>>>(lg, un, rmax, rsum, scal, h2);
  ppa_scan2_kernel<<<1, 1, 0, stream>>>(h2, scal);
  ppa_select_kernel<<<2048, 256, 0, stream>>>(lg, un, rmax, rsum, scal, bm, sel, eq);
  ppa_ties_kernel<<<1, 256, 0, stream>>>(scal, bm, sel, eq);
  ppa_render_kernel<<<KDOTS, 128, 0, stream>>>(sel, scal, bm, out);
}